// MemGuard_1589137899948
// MI455X (gfx1250) — compile-verified
//
#include <hip/hip_runtime.h>

// Row-wise argmax + constant fill ("MemGuard" defense) for logits[65536][1000] fp32.
//
// Memory-bound: 262 MB in + 262 MB out = 524 MB @ 23.3 TB/s => ~22.5 us floor.
// Strategy: one wave32 per row, b128 non-temporal streaming loads/stores,
// in-lane first-occurrence argmax + 5-step shfl_xor butterfly reduction.

typedef float v4f __attribute__((ext_vector_type(4)));

constexpr int C  = 1000;        // classes per row (fixed by the reference)
constexpr int V4 = C / 4;       // 250 float4 chunks per row (4000 B, 16B-aligned)
constexpr int WAVES_PER_BLOCK = 8;
constexpr int BLOCK = 32 * WAVES_PER_BLOCK;   // 256 threads = 8 wave32s

// Reference constants, computed in double then narrowed (matches Python/jnp):
//   on  = 1/1000 + 0.001 ; off = 1/1000 - 0.001/999
__device__ constexpr float kOn  = (float)(1.0 / (double)C + 0.001);
__device__ constexpr float kOff = (float)(1.0 / (double)C - 0.001 / (double)(C - 1));

__global__ __launch_bounds__(BLOCK) void memguard_argmax_fill(
    const float* __restrict__ logits, float* __restrict__ out, int rows) {
  const int lane = threadIdx.x & 31;                       // wave32 lane id
  const int row  = blockIdx.x * WAVES_PER_BLOCK + (threadIdx.x >> 5);
  if (row >= rows) return;

  const v4f* __restrict__ rin = reinterpret_cast<const v4f*>(logits + (size_t)row * C);

  // ---- per-lane argmax over its strided vec4 chunks (indices ascending) ----
  float best = -__builtin_inff();
  int   bidx = C;  // sentinel > any valid index

  #pragma unroll
  for (int k = 0; k < 8; ++k) {
    const int v = lane + 32 * k;        // vec4 index in row
    if (v < V4) {
      // streamed once: non-temporal load (th:NT) -> don't pollute L2
      const v4f d = __builtin_nontemporal_load(&rin[v]);
      const int base = 4 * v;
      // strict '>' while scanning ascending indices == first-occurrence argmax
      if (d.x > best) { best = d.x; bidx = base;     }
      if (d.y > best) { best = d.y; bidx = base + 1; }
      if (d.z > best) { best = d.z; bidx = base + 2; }
      if (d.w > best) { best = d.w; bidx = base + 3; }
    }
  }

  // ---- wave32 butterfly reduction; ties resolve toward the smaller index ----
  #pragma unroll
  for (int off = 16; off > 0; off >>= 1) {
    const float oval = __shfl_xor(best, off, 32);
    const int   oidx = __shfl_xor(bidx, off, 32);
    if (oval > best || (oval == best && oidx < bidx)) { best = oval; bidx = oidx; }
  }
  // all 32 lanes now agree on bidx (row argmax, first occurrence)

  // ---- constant fill, element-exact, b128 non-temporal stores ----
  v4f* __restrict__ rout = reinterpret_cast<v4f*>(out + (size_t)row * C);
  #pragma unroll
  for (int k = 0; k < 8; ++k) {
    const int v = lane + 32 * k;
    if (v < V4) {
      const int base = 4 * v;
      v4f o;
      o.x = (base     == bidx) ? kOn : kOff;
      o.y = (base + 1 == bidx) ? kOn : kOff;
      o.z = (base + 2 == bidx) ? kOn : kOff;
      o.w = (base + 3 == bidx) ? kOn : kOff;
      __builtin_nontemporal_store(o, &rout[v]);
    }
  }
}

extern "C" void kernel_launch(void* const* d_in, const int* in_sizes, int n_in,
                              void* d_out, int out_size, void* d_ws, size_t ws_size,
                              hipStream_t stream) {
  const float* logits = (const float*)d_in[0];
  float*       out    = (float*)d_out;

  const int rows   = in_sizes[0] / C;                       // 65536
  const int blocks = (rows + WAVES_PER_BLOCK - 1) / WAVES_PER_BLOCK;

  memguard_argmax_fill<<<blocks, BLOCK, 0, stream>>>(logits, out, rows);
}